// Neighbor_Context_57432302682834
// MI455X (gfx1250) — compile-verified
//
#include <hip/hip_runtime.h>
#include <hip/hip_bf16.h>

typedef __attribute__((ext_vector_type(2))) float v2f;
typedef __attribute__((ext_vector_type(8))) float v8f;

// Problem dims (fixed by the harness reference)
constexpr int B  = 4;
constexpr int C  = 128;     // channels
constexpr int NP = 4096;    // points
constexpr int K  = 32;      // neighbors (== wave32)
constexpr int NS = 16384;   // segment count N
constexpr int C2 = 2 * C;   // 256 concat channels

// Native CDNA5 fp32 max atomic (no return -> STOREcnt, no CAS loop).
// IEEE maximumNumber semantics match jnp segment_max composition exactly.
__device__ __forceinline__ void atomic_max_f32(float* addr, float v) {
    asm volatile("global_atomic_max_num_f32 %0, %1, off"
                 :
                 : "v"(addr), "v"(v)
                 : "memory");
}

// ---------------------------------------------------------------------------
// K0: zero the segment buffer (atomicMax accumulates, must re-zero per call)
// ---------------------------------------------------------------------------
__global__ __launch_bounds__(256) void k_zero(float* __restrict__ p, long nelem) {
    long i = (long)blockIdx.x * 256 + threadIdx.x;
    if (i < nelem) p[i] = 0.0f;
}

// ---------------------------------------------------------------------------
// K1: fused neighbor max-pool + scatter-max
//   block = one (b, n); 256 threads = 8 waves; lane = k neighbor.
//   Each wave handles one channel row per iteration (16 iters -> 128 ch).
//   x read once (coalesced 128B per wave); group_idx cached in LDS per block.
// ---------------------------------------------------------------------------
__global__ __launch_bounds__(256) void k_pool_scatter(
    const float* __restrict__ x, const int* __restrict__ gidx,
    float* __restrict__ seg, float* __restrict__ feat) {
    const int bn   = blockIdx.x;        // 0 .. B*NP-1
    const int b    = bn >> 12;          // NP = 4096
    const int n    = bn & (NP - 1);
    const int t    = threadIdx.x;
    const int lane = t & 31;
    const int wave = t >> 5;

    __shared__ int sidx[K];
    if (t < K) sidx[t] = gidx[(bn << 5) + t];
    __syncthreads();

    const int  myidx   = sidx[lane];
    const long segBase = ((long)b * NS + myidx) * C;

    for (int i = 0; i < 16; ++i) {
        const int   c = i * 8 + wave;
        const float v = x[(((long)(b * C + c)) * NP + n) * K + lane];
        // scatter-max into zero-initialized seg == segment_max then max(.,0)
        atomic_max_f32(&seg[segBase + c], v);
        // wave max over k for local pooling
        float m = v;
        #pragma unroll
        for (int off = 16; off; off >>= 1) m = fmaxf(m, __shfl_xor(m, off, 32));
        if (lane == 0) feat[(b * C2 + c) * NP + n] = m;  // group_x half
    }
}

// ---------------------------------------------------------------------------
// K2: gather dilated features: feat[b, C+c, n] = seg[b, fps[b,n], c]
//   n fastest -> coalesced writes; seg reads hit L2 (32MB buffer)
// ---------------------------------------------------------------------------
__global__ __launch_bounds__(256) void k_gather(
    const float* __restrict__ seg, const int* __restrict__ fps,
    float* __restrict__ feat) {
    const int tid = blockIdx.x * 256 + threadIdx.x;   // over B*C*NP
    const int n = tid & (NP - 1);
    const int c = (tid >> 12) & (C - 1);
    const int b = tid >> 19;
    const int f = fps[(b << 12) + n];
    feat[(b * C2 + C + c) * NP + n] = seg[((long)b * NS + f) * C + c];
}

// ---------------------------------------------------------------------------
// WMMA f32 16x16x4 fragment layouts (per ISA 7.12.2):
//   A (16x4): lane<16 -> M=lane, {K=k0,k0+1}; lane>=16 -> M=lane-16, {K=k0+2,k0+3}
//   B (4x16): VGPR0: K=k0 (lanes 0-15) / K=k0+2 (lanes 16-31); VGPR1: +1
//   C/D (16x16): VGPR v: M = v + 8*(lane>=16), N = lane&15
// ---------------------------------------------------------------------------

// K3: GEMM1  h1[b,o,n] = sum_c w1[o,c]*feat[b,c,n] + b1[o]
//   block = (b, 16-point tile); wave w owns M rows [16w, 16w+16)
__global__ __launch_bounds__(256) void k_gemm1(
    const float* __restrict__ w1, const float* __restrict__ b1,
    const float* __restrict__ feat, float* __restrict__ h1) {
    const int blk  = blockIdx.x;
    const int b    = blk >> 8;                 // NP/16 = 256 tiles per batch
    const int n0   = (blk & 255) * 16;
    const int t    = threadIdx.x;
    const int lane = t & 31;
    const int wave = t >> 5;
    const int half = lane >> 4;
    const int l16  = lane & 15;
    const int mrow = wave * 16 + l16;          // A-matrix row fed by this lane

    const float* fB = feat + (long)b * C2 * NP + n0 + l16;

    v8f acc = {};
    for (int k0 = 0; k0 < C2; k0 += 4) {
        const int ka = k0 + half * 2;
        v2f A, Bm;
        A.x  = w1[mrow * C2 + ka];
        A.y  = w1[mrow * C2 + ka + 1];
        Bm.x = fB[ka * NP];
        Bm.y = fB[(ka + 1) * NP];
        acc = __builtin_amdgcn_wmma_f32_16x16x4_f32(
            false, A, false, Bm, (short)0, acc, false, false);
    }
    #pragma unroll
    for (int v = 0; v < 8; ++v) {
        const int m = wave * 16 + v + 8 * half;
        h1[((long)(b * C + m)) * NP + n0 + l16] = acc[v] + b1[m];
    }
}

// ---------------------------------------------------------------------------
// K4: per-channel BN stats over (b, n); emits scale/shift so that
//     bn(x) = x*scale + shift  (ReLU applied by the consumer)
// ---------------------------------------------------------------------------
__global__ __launch_bounds__(256) void k_stats(
    const float* __restrict__ h, const float* __restrict__ g,
    const float* __restrict__ be, float* __restrict__ scale,
    float* __restrict__ shift) {
    const int ch = blockIdx.x;     // C blocks
    const int t  = threadIdx.x;    // 256
    float s = 0.f, s2 = 0.f;
    for (int b = 0; b < B; ++b) {
        const float* p = h + ((long)(b * C + ch)) * NP;
        for (int n = t; n < NP; n += 256) { float v = p[n]; s += v; s2 += v * v; }
    }
    #pragma unroll
    for (int off = 16; off; off >>= 1) {
        s  += __shfl_xor(s, off, 32);
        s2 += __shfl_xor(s2, off, 32);
    }
    __shared__ float rs[8], rs2[8];
    const int wave = t >> 5;
    if ((t & 31) == 0) { rs[wave] = s; rs2[wave] = s2; }
    __syncthreads();
    if (t == 0) {
        float S = 0.f, S2 = 0.f;
        #pragma unroll
        for (int w = 0; w < 8; ++w) { S += rs[w]; S2 += rs2[w]; }
        const float inv  = 1.0f / (float)(B * NP);
        const float mean = S * inv;
        const float var  = S2 * inv - mean * mean;
        const float sc   = g[ch] * rsqrtf(var + 1e-5f);
        scale[ch] = sc;
        shift[ch] = be[ch] - mean * sc;
    }
}

// ---------------------------------------------------------------------------
// K5: GEMM2 with BN1+ReLU fused into the B-operand load
//     out_raw[b,o,n] = sum_c w2[o,c]*relu(h1[b,c,n]*scale1[c]+shift1[c]) + b2[o]
// ---------------------------------------------------------------------------
__global__ __launch_bounds__(256) void k_gemm2(
    const float* __restrict__ w2, const float* __restrict__ b2,
    const float* __restrict__ h1, const float* __restrict__ scale1,
    const float* __restrict__ shift1, float* __restrict__ out) {
    const int blk  = blockIdx.x;
    const int b    = blk >> 8;
    const int n0   = (blk & 255) * 16;
    const int t    = threadIdx.x;
    const int lane = t & 31;
    const int wave = t >> 5;
    const int half = lane >> 4;
    const int l16  = lane & 15;
    const int mrow = wave * 16 + l16;

    const float* hB = h1 + (long)b * C * NP + n0 + l16;

    v8f acc = {};
    for (int k0 = 0; k0 < C; k0 += 4) {
        const int ka = k0 + half * 2;
        v2f A, Bm;
        A.x  = w2[mrow * C + ka];
        A.y  = w2[mrow * C + ka + 1];
        Bm.x = fmaxf(0.f, hB[ka * NP]       * scale1[ka]     + shift1[ka]);
        Bm.y = fmaxf(0.f, hB[(ka + 1) * NP] * scale1[ka + 1] + shift1[ka + 1]);
        acc = __builtin_amdgcn_wmma_f32_16x16x4_f32(
            false, A, false, Bm, (short)0, acc, false, false);
    }
    #pragma unroll
    for (int v = 0; v < 8; ++v) {
        const int m = wave * 16 + v + 8 * half;
        out[((long)(b * C + m)) * NP + n0 + l16] = acc[v] + b2[m];
    }
}

// ---------------------------------------------------------------------------
// K6: in-place BN2 + ReLU on d_out
// ---------------------------------------------------------------------------
__global__ __launch_bounds__(256) void k_bn_inplace(
    float* __restrict__ out, const float* __restrict__ sc,
    const float* __restrict__ sh) {
    const int i  = blockIdx.x * 256 + threadIdx.x;   // B*C*NP
    const int ch = (i >> 12) & (C - 1);
    out[i] = fmaxf(0.f, out[i] * sc[ch] + sh[ch]);
}

// ---------------------------------------------------------------------------
extern "C" void kernel_launch(void* const* d_in, const int* in_sizes, int n_in,
                              void* d_out, int out_size, void* d_ws, size_t ws_size,
                              hipStream_t stream) {
    const float* x    = (const float*)d_in[0];
    const int*   gidx = (const int*)d_in[1];
    const int*   fps  = (const int*)d_in[2];
    // d_in[3] = N (scalar, 16384) — compiled in
    const float* w1   = (const float*)d_in[4];
    const float* b1   = (const float*)d_in[5];
    const float* g1   = (const float*)d_in[6];
    const float* be1  = (const float*)d_in[7];
    const float* w2   = (const float*)d_in[8];
    const float* b2   = (const float*)d_in[9];
    const float* g2   = (const float*)d_in[10];
    const float* be2  = (const float*)d_in[11];
    float* out = (float*)d_out;

    // workspace layout (floats)
    float* ws     = (float*)d_ws;
    const long segN  = (long)B * NS * C;   // 8,388,608  (32 MB)
    const long featN = (long)B * C2 * NP;  // 4,194,304  (16 MB)
    const long h1N   = (long)B * C * NP;   // 2,097,152  ( 8 MB)
    float* seg    = ws;
    float* feat   = seg + segN;
    float* h1     = feat + featN;
    float* scale1 = h1 + h1N;
    float* shift1 = scale1 + C;
    float* scale2 = shift1 + C;
    float* shift2 = scale2 + C;

    k_zero<<<(int)((segN + 255) / 256), 256, 0, stream>>>(seg, segN);
    k_pool_scatter<<<B * NP, 256, 0, stream>>>(x, gidx, seg, feat);
    k_gather<<<(B * C * NP) / 256, 256, 0, stream>>>(seg, fps, feat);
    k_gemm1<<<B * (NP / 16), 256, 0, stream>>>(w1, b1, feat, h1);
    k_stats<<<C, 256, 0, stream>>>(h1, g1, be1, scale1, shift1);
    k_gemm2<<<B * (NP / 16), 256, 0, stream>>>(w2, b2, h1, scale1, shift1, out);
    k_stats<<<C, 256, 0, stream>>>(out, g2, be2, scale2, shift2);
    k_bn_inplace<<<(B * C * NP) / 256, 256, 0, stream>>>(out, scale2, shift2);
}